// decoder_bg_71983651881008
// MI455X (gfx1250) — compile-verified
//
#include <hip/hip_runtime.h>
#include <math.h>

typedef float v2f __attribute__((ext_vector_type(2)));
typedef float v4f __attribute__((ext_vector_type(4)));
typedef float v8f __attribute__((ext_vector_type(8)));

#define BN_EPS 1e-5f
#define THRESH 1e-5f

// ---------------------------------------------------------------------------
// Kernel 1: conv3x3(16->1) + BN + leakyReLU for both branches (f32 VALU),
// then the two (64x16)@(16x3) linears via V_WMMA_F32_16X16X4_F32 on wave 0,
// fused with sigmoid / affine / loss. Writes s[192], s_for[192], loss[192].
// ---------------------------------------------------------------------------
__global__ __launch_bounds__(256) void head_kernel(
    const float* __restrict__ x,
    const float* __restrict__ c1w, const float* __restrict__ c1b,
    const float* __restrict__ bn1g, const float* __restrict__ bn1b,
    const float* __restrict__ bn1m, const float* __restrict__ bn1v,
    const float* __restrict__ l1w, const float* __restrict__ l1b,
    const float* __restrict__ c2w, const float* __restrict__ c2b,
    const float* __restrict__ bn2g, const float* __restrict__ bn2b,
    const float* __restrict__ bn2m, const float* __restrict__ bn2v,
    const float* __restrict__ l2w, const float* __restrict__ l2b,
    float* __restrict__ out)
{
    __shared__ float hS[64 * 16];   // branch using x2 + conv1  -> s
    __shared__ float hF[64 * 16];   // branch using x1 + conv2  -> s_for

    const int t  = threadIdx.x;
    const int b  = t & 63;          // batch
    const int g  = t >> 6;          // 0..3
    const int br = g >> 1;          // 0: s-branch, 1: s_for-branch
    const int ph = g & 1;           // pixel half (8 pixels each)

    const float* w = (br == 0) ? c1w : c2w;
    const float cb = (br == 0) ? c1b[0] : c2b[0];
    const float bg = (br == 0) ? bn1g[0] : bn2g[0];
    const float bb = (br == 0) ? bn1b[0] : bn2b[0];
    const float bm = (br == 0) ? bn1m[0] : bn2m[0];
    const float bv = (br == 0) ? bn1v[0] : bn2v[0];
    const float inv = bg * rsqrtf(bv + BN_EPS);
    const float add = bb - bm * inv;
    const int cbase = (br == 0) ? 16 : 0;   // s-branch reads x[:,16:32]

    float acc[8];
    #pragma unroll
    for (int o = 0; o < 8; ++o) acc[o] = 0.f;

    const float* xb = x + ((size_t)b * 32 + cbase) * 16;
    for (int c = 0; c < 16; ++c) {
        const v4f* xp4 = (const v4f*)(xb + c * 16);
        v4f q0 = xp4[0], q1 = xp4[1], q2 = xp4[2], q3 = xp4[3];
        float xv[16] = {q0.x, q0.y, q0.z, q0.w, q1.x, q1.y, q1.z, q1.w,
                        q2.x, q2.y, q2.z, q2.w, q3.x, q3.y, q3.z, q3.w};
        const float* wc = w + c * 9;
        float w9[9];
        #pragma unroll
        for (int k = 0; k < 9; ++k) w9[k] = wc[k];
        #pragma unroll
        for (int o = 0; o < 8; ++o) {
            const int p  = ph * 8 + o;
            const int oh = p >> 2, ow = p & 3;
            #pragma unroll
            for (int kh = 0; kh < 3; ++kh) {
                const int r = oh + kh - 1;
                if (r < 0 || r > 3) continue;
                #pragma unroll
                for (int kw = 0; kw < 3; ++kw) {
                    const int cc = ow + kw - 1;
                    if (cc < 0 || cc > 3) continue;
                    acc[o] = fmaf(xv[r * 4 + cc], w9[kh * 3 + kw], acc[o]);
                }
            }
        }
    }
    float* hdst = (br == 0) ? hS : hF;
    #pragma unroll
    for (int o = 0; o < 8; ++o) {
        float v = fmaf(acc[o] + cb, inv, add);   // (conv+b)*inv + (beta-mean*inv)
        v = (v > 0.f) ? v : 0.2f * v;            // leaky relu 0.2
        hdst[b * 16 + ph * 8 + o] = v;
    }
    __syncthreads();

    // ------ wave 0: linear layers as WMMA f32 16x16x4, K=16 in 4 steps ------
    if (t < 32) {
        const int half = t >> 4;    // 0: lanes 0-15, 1: lanes 16-31
        const int n    = t & 15;    // output column (only 0..2 meaningful)

        // B (4x16) per K-step: VGPR0 holds K=0(+2*half), VGPR1 holds K=1(+2*half)
        v2f bS[4], bF[4];
        #pragma unroll
        for (int ks = 0; ks < 4; ++ks) {
            const int k0 = ks * 4 + half * 2;
            if (n < 3) {
                bS[ks].x = l1w[n * 16 + k0];  bS[ks].y = l1w[n * 16 + k0 + 1];
                bF[ks].x = l2w[n * 16 + k0];  bF[ks].y = l2w[n * 16 + k0 + 1];
            } else {
                bS[ks].x = 0.f; bS[ks].y = 0.f;
                bF[ks].x = 0.f; bF[ks].y = 0.f;
            }
        }
        const float lbS = (n < 3) ? l1b[n] : 0.f;
        const float lbF = (n < 3) ? l2b[n] : 0.f;

        for (int mt = 0; mt < 4; ++mt) {        // 4 tiles of 16 batches
            v8f cS = {};
            v8f cF = {};
            #pragma unroll
            for (int ks = 0; ks < 4; ++ks) {
                const int M  = mt * 16 + n;      // A row = lane&15
                const int kk = ks * 4 + half * 2;
                v2f aS, aF;
                aS.x = hS[M * 16 + kk]; aS.y = hS[M * 16 + kk + 1];
                aF.x = hF[M * 16 + kk]; aF.y = hF[M * 16 + kk + 1];
                cS = __builtin_amdgcn_wmma_f32_16x16x4_f32(
                        false, aS, false, bS[ks], (short)0, cS, false, false);
                cF = __builtin_amdgcn_wmma_f32_16x16x4_f32(
                        false, aF, false, bF[ks], (short)0, cF, false, false);
            }
            if (n < 3) {
                #pragma unroll
                for (int r = 0; r < 8; ++r) {
                    const int M = mt * 16 + r + half * 8;   // D row per §7.12.2
                    const float s  = 1.f / (1.f + expf(-(cS[r] + lbS)));
                    const float sf = (1.f / (1.f + expf(-(cF[r] + lbF))) + 0.125f) * 0.8f;
                    const float d    = fabsf(s - sf);
                    const float mask = (d <= THRESH) ? 1.f : 0.f;
                    out[        M * 3 + n] = s;
                    out[192   + M * 3 + n] = sf;
                    out[384   + M * 3 + n] = 1000.f * mask / d;
                }
            }
        }
    }
}

// ---------------------------------------------------------------------------
// Kernel 2: col = where(orig==0, s_bcast, where(orig==1, sf_bcast, orig)).
// One (b,c) plane per blockIdx.y -> sb/sf are block-uniform scalars.
// 402 MB of streaming traffic > 192 MB L2, so use non-temporal B128 ld/st.
// ---------------------------------------------------------------------------
__global__ __launch_bounds__(256) void col_kernel(
    const float* __restrict__ orig, const float* __restrict__ s,
    const float* __restrict__ sfor, float* __restrict__ col)
{
    const int plane = blockIdx.y;           // b*3 + c, 0..191
    const float sb = s[plane];
    const float sf = sfor[plane];
    const size_t base4 = (size_t)plane * 65536;   // 512*512/4 float4 per plane
    const v4f* in  = (const v4f*)orig + base4;
    v4f*       dst = (v4f*)col + base4;

    const int t = blockIdx.x * 256 + threadIdx.x;  // 0..16383 within plane
    #pragma unroll
    for (int i = 0; i < 4; ++i) {
        const int idx = t + i * 16384;
        v4f v = __builtin_nontemporal_load(&in[idx]);
        v.x = (v.x == 0.f) ? sb : ((v.x == 1.f) ? sf : v.x);
        v.y = (v.y == 0.f) ? sb : ((v.y == 1.f) ? sf : v.y);
        v.z = (v.z == 0.f) ? sb : ((v.z == 1.f) ? sf : v.z);
        v.w = (v.w == 0.f) ? sb : ((v.w == 1.f) ? sf : v.w);
        __builtin_nontemporal_store(v, &dst[idx]);
    }
}

extern "C" void kernel_launch(void* const* d_in, const int* in_sizes, int n_in,
                              void* d_out, int out_size, void* d_ws, size_t ws_size,
                              hipStream_t stream) {
    const float* x    = (const float*)d_in[0];
    const float* orig = (const float*)d_in[1];
    const float* c1w  = (const float*)d_in[2];
    const float* c1b  = (const float*)d_in[3];
    const float* bn1g = (const float*)d_in[4];
    const float* bn1b = (const float*)d_in[5];
    const float* bn1m = (const float*)d_in[6];
    const float* bn1v = (const float*)d_in[7];
    const float* l1w  = (const float*)d_in[8];
    const float* l1b  = (const float*)d_in[9];
    const float* c2w  = (const float*)d_in[10];
    const float* c2b  = (const float*)d_in[11];
    const float* bn2g = (const float*)d_in[12];
    const float* bn2b = (const float*)d_in[13];
    const float* bn2m = (const float*)d_in[14];
    const float* bn2v = (const float*)d_in[15];
    const float* l2w  = (const float*)d_in[16];
    const float* l2b  = (const float*)d_in[17];
    float* out = (float*)d_out;   // [s:192][s_for:192][loss:192][col:50331648]

    head_kernel<<<1, 256, 0, stream>>>(x, c1w, c1b, bn1g, bn1b, bn1m, bn1v,
                                       l1w, l1b, c2w, c2b, bn2g, bn2b, bn2m,
                                       bn2v, l2w, l2b, out);

    dim3 grid(64, 192);
    col_kernel<<<grid, 256, 0, stream>>>(orig, out, out + 192, out + 576);
}